// GatedAttentionAggregator_35742717837626
// MI455X (gfx1250) — compile-verified
//
#include <hip/hip_runtime.h>
#include <hip/hip_bf16.h>

typedef __attribute__((ext_vector_type(2))) float v2f;
typedef __attribute__((ext_vector_type(8))) float v8f;

#define GDIN   128
#define GDOUT  128
#define GH     8
#define GDH    16
#define GDG    64
#define GCOLS  320           // 128 (order0) + 128 (order1) + 64 (pool gate)
#define NEG_SLOPE 0.2f

// ---- fp32 scatter-add that lowers to native global_atomic_add_f32 on CDNA5 ----
__device__ __forceinline__ void atomAddF(float* p, float v) {
    unsafeAtomicAdd(p, v);
}

// ---------------- order-preserving float <-> uint key (for atomic segment max) ----
__device__ __forceinline__ unsigned fkey(float f) {
    unsigned u = __float_as_uint(f);
    return (u & 0x80000000u) ? ~u : (u | 0x80000000u);
}
__device__ __forceinline__ float fdecode(unsigned k) {
    unsigned u = (k & 0x80000000u) ? (k & 0x7FFFFFFFu) : ~k;
    return __uint_as_float(u);
}

// ---------------- repack W[2,H,DIN,DH] + Wpool[DIN,DG] into Bbig[DIN, 320] ---------
__global__ void k_repack(const float* __restrict__ W, const float* __restrict__ Wpool,
                         float* __restrict__ Bbig) {
    int idx = blockIdx.x * blockDim.x + threadIdx.x;
    if (idx >= GDIN * GCOLS) return;
    int d = idx / GCOLS, c = idx % GCOLS;
    float v;
    if (c < 256) {
        int o = c >> 7, h = (c & 127) >> 4, k = c & 15;
        v = W[(((size_t)o * GH + h) * GDIN + d) * GDH + k];
    } else {
        v = Wpool[(size_t)d * GDG + (c - 256)];
    }
    Bbig[idx] = v;
}

// ---------------- fused GEMM: P = feat @ Bbig ; relu(P[:, :256]+bias) -> f0,f1 ; P[:,256:] -> zj
// One wave computes a 16-row x 64-col strip using V_WMMA_F32_16X16X4_F32.
__global__ void k_gemm_feat(const float* __restrict__ feat, const float* __restrict__ Bbig,
                            const float* __restrict__ bias,        // [2,H,DH] = [2][128]
                            float* __restrict__ f0, float* __restrict__ f1,
                            float* __restrict__ zj, int rowTiles, int Nn) {
    const int lane = threadIdx.x & 31;
    const int wave = threadIdx.x >> 5;                 // 4 waves / block
    const int rowTile = blockIdx.x * 4 + wave;
    if (rowTile >= rowTiles) return;                   // uniform across the wave
    const int colBase = blockIdx.y * 64;               // 5 column groups
    const int row0 = rowTile * 16;

    const int mA   = lane & 15;                        // M (lanes 0-15) / M (lanes 16-31)
    const int kOff = (lane >> 4) * 2;                  // lanes>=16 hold K+2,K+3

    v8f acc0 = {}; v8f acc1 = {}; v8f acc2 = {}; v8f acc3 = {};

    for (int k = 0; k < GDIN; k += 4) {
        // A fragment (16x4 fp32): row mA, columns k+kOff, k+kOff+1 (contiguous -> b64 load)
        const float* ap = feat + (size_t)(row0 + mA) * GDIN + (k + kOff);
        v2f a; a[0] = ap[0]; a[1] = ap[1];
        // B fragments (4x16 fp32, mirrored layout): row k+kOff(+1), col colBase+t*16+mA
        const float* bp = Bbig + (size_t)(k + kOff) * GCOLS + colBase + mA;
        v2f b0, b1, b2, b3;
        b0[0] = bp[0];  b0[1] = bp[GCOLS];
        b1[0] = bp[16]; b1[1] = bp[GCOLS + 16];
        b2[0] = bp[32]; b2[1] = bp[GCOLS + 32];
        b3[0] = bp[48]; b3[1] = bp[GCOLS + 48];
        acc0 = __builtin_amdgcn_wmma_f32_16x16x4_f32(false, a, false, b0, (short)0, acc0, false, false);
        acc1 = __builtin_amdgcn_wmma_f32_16x16x4_f32(false, a, false, b1, (short)0, acc1, false, false);
        acc2 = __builtin_amdgcn_wmma_f32_16x16x4_f32(false, a, false, b2, (short)0, acc2, false, false);
        acc3 = __builtin_amdgcn_wmma_f32_16x16x4_f32(false, a, false, b3, (short)0, acc3, false, false);
    }

    // Epilogue. C/D layout: VGPR i -> row M = i (lanes 0-15) or 8+i (lanes 16-31); col N = lane%16.
    const int nC    = lane & 15;
    const int mHalf = (lane >> 4) * 8;
    #pragma unroll
    for (int i = 0; i < 8; ++i) {
        const int r = row0 + mHalf + i;
        if (r >= Nn) continue;
        float vv[4] = { acc0[i], acc1[i], acc2[i], acc3[i] };
        #pragma unroll
        for (int t = 0; t < 4; ++t) {
            const int c = colBase + t * 16 + nC;
            float v = vv[t];
            if (c < 256) {
                const int o  = c >> 7;
                const int cc = c & 127;                // == h*16 + kdim
                v += bias[o * 128 + cc];
                v = v > 0.0f ? v : 0.0f;               // relu
                (o == 0 ? f0 : f1)[(size_t)r * GDOUT + cc] = v;
            } else {
                zj[(size_t)r * GDG + (c - 256)] = v;
            }
        }
    }
}

// ---------------- attention coefficients: leaky(f{0,1}[n,h,:] . att[h, :16 / 16:]) ----
__global__ void k_attn(const float* __restrict__ f0, const float* __restrict__ f1,
                       const float* __restrict__ att,
                       float* __restrict__ a_self, float* __restrict__ a_neigh, int Nn) {
    int idx = blockIdx.x * blockDim.x + threadIdx.x;
    if (idx >= Nn * GH) return;
    int n = idx >> 3, h = idx & 7;
    float s0 = 0.f, s1 = 0.f;
    #pragma unroll
    for (int k = 0; k < GDH; ++k) {
        s0 += f0[(size_t)n * GDOUT + h * GDH + k] * att[h * 2 * GDH + k];
        s1 += f1[(size_t)n * GDOUT + h * GDH + k] * att[h * 2 * GDH + GDH + k];
    }
    a_self[idx]  = s0 >= 0.f ? s0 : NEG_SLOPE * s0;
    a_neigh[idx] = s1 >= 0.f ? s1 : NEG_SLOPE * s1;
}

// ---------------- edge pass 1a: segment max of zj[col] into zkey[row] ----------------
// One thread per (edge, 4-dim group): b128 gather + 4 uint atomicMax.
__global__ void k_edge_zmax(const int* __restrict__ row, const int* __restrict__ col,
                            const float* __restrict__ zj, unsigned* __restrict__ zkey,
                            long long total) {
    long long idx = (long long)blockIdx.x * blockDim.x + threadIdx.x;
    if (idx >= total) return;                          // total = E * (GDG/4)
    int e = (int)(idx >> 4), g = (int)(idx & 15);
    int d0 = g * 4;
    const float4 z = *(const float4*)(zj + (size_t)col[e] * GDG + d0);
    unsigned* dst = zkey + (size_t)row[e] * GDG + d0;
    atomicMax(dst + 0, fkey(z.x));
    atomicMax(dst + 1, fkey(z.y));
    atomicMax(dst + 2, fkey(z.z));
    atomicMax(dst + 3, fkey(z.w));
}

// ---------------- edge pass 1b: nmean[row] += val * feat[col] ------------------------
// One thread per (edge, 4-dim group): b128 gather + 4 native f32 atomic adds.
__global__ void k_edge_nmean(const int* __restrict__ row, const int* __restrict__ col,
                             const float* __restrict__ val, const float* __restrict__ feat,
                             float* __restrict__ nmean, long long total) {
    long long idx = (long long)blockIdx.x * blockDim.x + threadIdx.x;
    if (idx >= total) return;                          // total = E * (GDIN/4)
    int e = (int)(idx >> 5), g = (int)(idx & 31);
    int d0 = g * 4;
    const float w = val[e];
    const float4 f = *(const float4*)(feat + (size_t)col[e] * GDIN + d0);
    float* dst = nmean + (size_t)row[e] * GDIN + d0;
    atomAddF(dst + 0, w * f.x);
    atomAddF(dst + 1, w * f.y);
    atomAddF(dst + 2, w * f.z);
    atomAddF(dst + 3, w * f.w);
}

// ---------------- gate = [feat | zmax | nmean] @ weight_gate  ([N,320]x[320,8]) ------
__global__ void k_gate(const float* __restrict__ feat, const unsigned* __restrict__ zkey,
                       const float* __restrict__ nmean, const float* __restrict__ wg,
                       float* __restrict__ gate, int Nn) {
    int idx = blockIdx.x * blockDim.x + threadIdx.x;
    if (idx >= Nn * GH) return;
    int n = idx >> 3, h = idx & 7;
    float s = 0.f;
    for (int i = 0; i < GDIN; ++i) s += feat[(size_t)n * GDIN + i] * wg[i * GH + h];
    for (int i = 0; i < GDG; ++i) {
        unsigned k = zkey[(size_t)n * GDG + i];
        float z = (k == 0u) ? 0.0f : fdecode(k);       // untouched -> isolated node -> 0
        s += z * wg[(GDIN + i) * GH + h];
    }
    for (int i = 0; i < GDIN; ++i) s += nmean[(size_t)n * GDIN + i] * wg[(GDIN + GDG + i) * GH + h];
    gate[idx] = s;
}

// ---------------- edge pass 2: agg[row] += ((a_self[row]+a_neigh[col])*val) * f1[col] -
// One thread per (edge, 4-dim group); the 4 dims share one head (16/4).
__global__ void k_edge_agg(const int* __restrict__ row, const int* __restrict__ col,
                           const float* __restrict__ val, const float* __restrict__ f1,
                           const float* __restrict__ a_self, const float* __restrict__ a_neigh,
                           float* __restrict__ agg, long long total) {
    long long idx = (long long)blockIdx.x * blockDim.x + threadIdx.x;
    if (idx >= total) return;                          // total = E * (GDOUT/4)
    int e = (int)(idx >> 5), g = (int)(idx & 31);
    int c0 = g * 4, h = c0 >> 4;
    int r = row[e], cl = col[e];
    float w = (a_self[(size_t)r * GH + h] + a_neigh[(size_t)cl * GH + h]) * val[e];
    const float4 f = *(const float4*)(f1 + (size_t)cl * GDOUT + c0);
    float* dst = agg + (size_t)r * GDOUT + c0;
    atomAddF(dst + 0, w * f.x);
    atomAddF(dst + 1, w * f.y);
    atomAddF(dst + 2, w * f.z);
    atomAddF(dst + 3, w * f.w);
}

// ---------------- per-node norm + sum. One wave32 per node, shfl reductions ----------
__global__ void k_final(const float* __restrict__ f0, const float* __restrict__ agg,
                        const float* __restrict__ gate, const float* __restrict__ scale,
                        const float* __restrict__ offset, float* __restrict__ out, int Nn) {
    const int lane = threadIdx.x & 31;
    const int wave = threadIdx.x >> 5;
    const int n = blockIdx.x * (blockDim.x >> 5) + wave;
    if (n >= Nn) return;
    float x0[4], x1[4];
    float s0 = 0.f, q0 = 0.f, s1 = 0.f, q1 = 0.f;
    #pragma unroll
    for (int j = 0; j < 4; ++j) {
        const int c = lane + 32 * j;
        float a = f0[(size_t)n * GDOUT + c];
        float b = agg[(size_t)n * GDOUT + c] * gate[(size_t)n * GH + (c >> 4)];
        x0[j] = a; x1[j] = b;
        s0 += a; q0 += a * a; s1 += b; q1 += b * b;
    }
    #pragma unroll
    for (int m = 16; m >= 1; m >>= 1) {
        s0 += __shfl_xor(s0, m, 32); q0 += __shfl_xor(q0, m, 32);
        s1 += __shfl_xor(s1, m, 32); q1 += __shfl_xor(q1, m, 32);
    }
    const float inv = 1.0f / (float)GDOUT;
    float mu0 = s0 * inv, mu1 = s1 * inv;
    float v0 = q0 * inv - mu0 * mu0 + 1e-9f;
    float v1 = q1 * inv - mu1 * mu1 + 1e-9f;
    float r0 = rsqrtf(v0), r1 = rsqrtf(v1);
    #pragma unroll
    for (int j = 0; j < 4; ++j) {
        const int c = lane + 32 * j;
        float h0 = (x0[j] - mu0) * scale[c]       * r0 + offset[c];
        float h1 = (x1[j] - mu1) * scale[128 + c] * r1 + offset[128 + c];
        out[(size_t)n * GDOUT + c] = h0 + h1;
    }
}

extern "C" void kernel_launch(void* const* d_in, const int* in_sizes, int n_in,
                              void* d_out, int out_size, void* d_ws, size_t ws_size,
                              hipStream_t stream) {
    const int*   row   = (const int*)  d_in[0];
    const int*   col   = (const int*)  d_in[1];
    const float* val   = (const float*)d_in[2];
    const float* feat  = (const float*)d_in[3];
    const float* W     = (const float*)d_in[4];
    const float* bias  = (const float*)d_in[5];
    const float* att   = (const float*)d_in[6];
    const float* off   = (const float*)d_in[7];
    const float* scl   = (const float*)d_in[8];
    const float* wg    = (const float*)d_in[9];
    const float* wpool = (const float*)d_in[10];

    const int E  = in_sizes[0];
    const int Nn = in_sizes[3] / GDIN;

    // ---- workspace carve-up (floats; every block a multiple of 16B) ----
    float* ws = (float*)d_ws;
    float*    Bbig   = ws;                 ws += GDIN * GCOLS;
    float*    f0     = ws;                 ws += (size_t)Nn * GDOUT;
    float*    f1     = ws;                 ws += (size_t)Nn * GDOUT;
    float*    zj     = ws;                 ws += (size_t)Nn * GDG;
    unsigned* zkey   = (unsigned*)ws;      ws += (size_t)Nn * GDG;
    float*    nmean  = ws;                 ws += (size_t)Nn * GDIN;
    float*    agg    = ws;                 ws += (size_t)Nn * GDOUT;
    float*    a_self = ws;                 ws += (size_t)Nn * GH;
    float*    a_nei  = ws;                 ws += (size_t)Nn * GH;
    float*    gate   = ws;                 ws += (size_t)Nn * GH;

    // deterministic re-init of atomic destinations (capturable memset nodes)
    hipMemsetAsync(zkey,  0, (size_t)Nn * GDG  * sizeof(unsigned), stream);
    hipMemsetAsync(nmean, 0, (size_t)Nn * GDIN * sizeof(float),    stream);
    hipMemsetAsync(agg,   0, (size_t)Nn * GDOUT* sizeof(float),    stream);

    // 0) repack weights
    k_repack<<<(GDIN * GCOLS + 255) / 256, 256, 0, stream>>>(W, wpool, Bbig);

    // 1) fused WMMA GEMM -> f0, f1 (relu+bias), zj
    const int rowTiles = (Nn + 15) / 16;   // N=50000 -> 3125 exact
    dim3 gg((rowTiles + 3) / 4, GCOLS / 64);
    k_gemm_feat<<<gg, 128, 0, stream>>>(feat, Bbig, bias, f0, f1, zj, rowTiles, Nn);

    // 2) attention coefficients
    k_attn<<<(Nn * GH + 255) / 256, 256, 0, stream>>>(f0, f1, att, a_self, a_nei, Nn);

    // 3) edge passes (L2-resident gather + native atomic scatter)
    {
        long long t1 = (long long)E * (GDG / 4);
        k_edge_zmax<<<(unsigned)((t1 + 255) / 256), 256, 0, stream>>>(row, col, zj, zkey, t1);
        long long t2 = (long long)E * (GDIN / 4);
        k_edge_nmean<<<(unsigned)((t2 + 255) / 256), 256, 0, stream>>>(row, col, val, feat, nmean, t2);
    }

    // 4) gate GEMV
    k_gate<<<(Nn * GH + 255) / 256, 256, 0, stream>>>(feat, zkey, nmean, wg, gate, Nn);

    // 5) attention-weighted aggregation
    {
        long long t3 = (long long)E * (GDOUT / 4);
        k_edge_agg<<<(unsigned)((t3 + 255) / 256), 256, 0, stream>>>(row, col, val, f1, a_self, a_nei, agg, t3);
    }

    // 6) per-node norm + combine
    k_final<<<(Nn + 7) / 8, 256, 0, stream>>>(f0, agg, gate, scl, off, (float*)d_out, Nn);
}